// GNNMultiClass_52673478918387
// MI455X (gfx1250) — compile-verified
//
#include <hip/hip_runtime.h>

// Problem constants (B=1)
#define NNODES 8192   // N
#define NFEAT  32     // F
#define NJ     4      // J
#define NCLS   10     // classes
#define OPAD   16     // classes padded to WMMA M

typedef __attribute__((ext_vector_type(2))) float v2f;
typedef __attribute__((ext_vector_type(8))) float v8f;

// -----------------------------------------------------------------------------
// Kernel 1: u[n][o][j] = sum_f conv_w[o, j*32+f] * x[f, n]   (o>=10 -> 0)
// u layout: [NNODES][OPAD][NJ] floats (2 MB), the A-matrix feed for WMMA.
// -----------------------------------------------------------------------------
__global__ void build_u_kernel(const float* __restrict__ x,
                               const float* __restrict__ cw,
                               float* __restrict__ u) {
    int idx = blockIdx.x * blockDim.x + threadIdx.x;   // NNODES*OPAD*NJ threads
    int n = idx >> 6;            // /64
    int o = (idx >> 2) & 15;
    int j = idx & 3;
    float s = 0.0f;
    if (o < NCLS) {
        const float* xr = x + n;                       // x[f*N + n]
        const float* wr = cw + o * (NJ * NFEAT) + j * NFEAT;
        #pragma unroll
        for (int f = 0; f < NFEAT; ++f)
            s += wr[f] * xr[(size_t)f * NNODES];
    }
    u[idx] = s;
}

// -----------------------------------------------------------------------------
// Kernel 2: main GEMM via V_WMMA_F32_16X16X4_F32.
//   part[kc][m][o] = sum_{n in chunk kc} sum_j u[o,j,n] * W[n,m,j]
// One wave per (m-tile of 16, k-chunk). K dim of each WMMA = j (contiguous in W).
// W is streamed once (1 GiB >> 192 MB L2) -> non-temporal loads so the hot,
// shared u table (2 MB) stays L2-resident.
// -----------------------------------------------------------------------------
__global__ void __launch_bounds__(256)
gmul_wmma_kernel(const float* __restrict__ W,
                 const float* __restrict__ u,
                 float* __restrict__ part,
                 int kch, int chunk) {
    const int lane  = threadIdx.x & 31;
    const int wid   = (blockIdx.x * blockDim.x + threadIdx.x) >> 5;
    const int mtile = wid / kch;
    const int kc    = wid - mtile * kch;
    const int col   = lane & 15;     // B column (m) / A row (o)
    const int kh    = lane >> 4;     // K half: j pair {0,1} or {2,3}
    const int m0    = mtile * 16;

    const int nb = kc * chunk;
    // B feed: W[n][m0+col][2kh .. 2kh+1]  (lanes cover 256 contiguous bytes)
    const float* bp = W + (size_t)nb * (NNODES * NJ)
                        + (size_t)(m0 + col) * NJ + kh * 2;
    // A feed: u[n][col][2kh .. 2kh+1]
    const float* ap = u + (size_t)nb * (OPAD * NJ) + col * NJ + kh * 2;

    v8f c0 = {}, c1 = {}, c2 = {}, c3 = {};
    #pragma unroll 2
    for (int it = 0; it < chunk; it += 4) {
        v2f a0 = *(const v2f*)(ap);
        v2f a1 = *(const v2f*)(ap + 1 * (OPAD * NJ));
        v2f a2 = *(const v2f*)(ap + 2 * (OPAD * NJ));
        v2f a3 = *(const v2f*)(ap + 3 * (OPAD * NJ));
        v2f b0 = __builtin_nontemporal_load((const v2f*)(bp));
        v2f b1 = __builtin_nontemporal_load((const v2f*)(bp + 1 * (size_t)(NNODES * NJ)));
        v2f b2 = __builtin_nontemporal_load((const v2f*)(bp + 2 * (size_t)(NNODES * NJ)));
        v2f b3 = __builtin_nontemporal_load((const v2f*)(bp + 3 * (size_t)(NNODES * NJ)));
        // 8 args: (neg_a, A, neg_b, B, c_mod, C, reuse_a, reuse_b)
        c0 = __builtin_amdgcn_wmma_f32_16x16x4_f32(false, a0, false, b0,
                                                   (short)0, c0, false, false);
        c1 = __builtin_amdgcn_wmma_f32_16x16x4_f32(false, a1, false, b1,
                                                   (short)0, c1, false, false);
        c2 = __builtin_amdgcn_wmma_f32_16x16x4_f32(false, a2, false, b2,
                                                   (short)0, c2, false, false);
        c3 = __builtin_amdgcn_wmma_f32_16x16x4_f32(false, a3, false, b3,
                                                   (short)0, c3, false, false);
        ap += 4 * (OPAD * NJ);
        bp += 4 * (size_t)(NNODES * NJ);
    }
    v8f c = (c0 + c1) + (c2 + c3);

    // D layout: lane holds rows (kh*8 .. kh*8+7) of column (m0+col) -> one 32B store
    float* p = part + (size_t)kc * (NNODES * OPAD)
                    + (size_t)(m0 + col) * OPAD + kh * 8;
    *(v8f*)p = c;
}

// -----------------------------------------------------------------------------
// Kernel 3: deterministic reduction over k-chunks + bias, write out[m*10+o].
// -----------------------------------------------------------------------------
__global__ void finalize_kernel(const float* __restrict__ part,
                                const float* __restrict__ bias,
                                float* __restrict__ out, int kch) {
    int idx = blockIdx.x * blockDim.x + threadIdx.x;
    if (idx >= NNODES * NCLS) return;
    int m = idx / NCLS;
    int o = idx - m * NCLS;
    float s = bias[o];
    for (int k = 0; k < kch; ++k)
        s += part[(size_t)k * (NNODES * OPAD) + m * OPAD + o];
    out[idx] = s;
}

extern "C" void kernel_launch(void* const* d_in, const int* in_sizes, int n_in,
                              void* d_out, int out_size, void* d_ws, size_t ws_size,
                              hipStream_t stream) {
    const float* x  = (const float*)d_in[0];   // [1,32,8192,1]
    const float* W  = (const float*)d_in[1];   // [1,8192,8192,4]
    const float* cw = (const float*)d_in[2];   // [10,128,1,1]
    const float* cb = (const float*)d_in[3];   // [10]
    float* out = (float*)d_out;                // [8192,10]

    // Workspace layout: u (2 MB) | partials (kch * 512 KB)
    const size_t u_bytes = (size_t)NNODES * OPAD * NJ * sizeof(float);
    int kch = 8;
    while (kch > 1 &&
           u_bytes + (size_t)kch * NNODES * OPAD * sizeof(float) > ws_size)
        kch >>= 1;
    float* u    = (float*)d_ws;
    float* part = (float*)((char*)d_ws + u_bytes);
    const int chunk = NNODES / kch;   // multiple of 4, >= 1024

    // 1) tiny channel-fold GEMM: u = conv_w * x
    build_u_kernel<<<(NNODES * OPAD * NJ) / 256, 256, 0, stream>>>(x, cw, u);

    // 2) main fp32 WMMA GEMM: 512 m-tiles * kch chunks waves, 8 waves/block
    const int waves  = (NNODES / 16) * kch;
    gmul_wmma_kernel<<<waves / 8, 256, 0, stream>>>(W, u, part, kch, chunk);

    // 3) reduce partials + bias -> out
    finalize_kernel<<<(NNODES * NCLS + 255) / 256, 256, 0, stream>>>(part, cb, out, kch);
}